// Enc_apg_z_33663953666955
// MI455X (gfx1250) — compile-verified
//
#include <hip/hip_runtime.h>

// Problem constants (from reference): S,B,N,K,D,H
#define SS 4
#define BB 8
#define NN 2048
#define KK 16
#define DD 32
#define HH 128
#define SB (SS * BB)                 // 32
#define TPB 128                      // 4 waves (wave32)
#define M_TILE 128                   // rows of N per workgroup
#define TILES_PER_SB (NN / M_TILE)   // 16
#define NUM_WG (SB * TILES_PER_SB)   // 512

// LDS layout (floats). Padded strides for bank-conflict-free access.
// ldsHX is ALIASED over W1 rows 32..95 (Wm/Wt), which are dead after the
// hk phase — a barrier separates last Wm/Wt read from first HX write.
#define W1_STRIDE 130
#define HX_STRIDE 130                // even -> 8B-aligned rows; 130%64=2 ->
                                     // b64 reads across 32 lanes hit all 64 banks once
#define OFF_W1 0                     // [96][W1_STRIDE]; rows 0..31 = Wx persist
#define OFF_HX (32 * W1_STRIDE)      // 4160; overlays rows 32..95 of W1
#define OFF_HK (OFF_HX + M_TILE * HX_STRIDE)   // 4160 + 16640 = 20800
#define OFF_MT (OFF_HK + KK * HH)              // +2048 -> 22848
#define OFF_W2 (OFF_MT + 2 * KK * DD)          // +1024 -> 23872
#define LDS_FLOATS (OFF_W2 + HH)               // +128  -> 24000 floats (~94 KB)

typedef __attribute__((ext_vector_type(2))) float v2f;
typedef __attribute__((ext_vector_type(4))) float v4f;
typedef __attribute__((ext_vector_type(8))) float v8f;

__device__ __forceinline__ float fast_tanh(float v) {
#if __has_builtin(__builtin_amdgcn_tanhf)
  return __builtin_amdgcn_tanhf(v);        // v_tanh_f32 (gfx1250 trans op)
#elif __has_builtin(__builtin_amdgcn_tanh_f32)
  return __builtin_amdgcn_tanh_f32(v);
#else
  float e = __expf(-2.0f * v);             // fallback via v_exp_f32
  return (1.0f - e) / (1.0f + e);
#endif
}

__global__ __launch_bounds__(TPB)
void apg_fused_kernel(const float* __restrict__ x,
                      const float* __restrict__ mu,
                      const float* __restrict__ tau,
                      const float* __restrict__ W1,
                      const float* __restrict__ b1,
                      const float* __restrict__ W2,
                      float* __restrict__ out) {
  extern __shared__ float smem[];
  float* ldsW1 = smem + OFF_W1;   // [96][W1_STRIDE]: Wx(0..31) Wm(32..63) Wt(64..95)
  float* ldsHX = smem + OFF_HX;   // [M_TILE][HX_STRIDE] (aliases Wm/Wt after barrier)
  float* ldsHK = smem + OFF_HK;   // [KK][HH]  (includes b1)
  float* ldsMT = smem + OFF_MT;   // mu tile [KK][DD] then tau tile [KK][DD]
  float* ldsW2 = smem + OFF_W2;   // [HH]

  const int tid = threadIdx.x;
  const int sb = blockIdx.x / TILES_PER_SB;      // 0..31
  const int tile = blockIdx.x % TILES_PER_SB;    // 0..15
  const int rowBase = tile * M_TILE;

  const float* xsb  = x   + (size_t)sb * NN * DD;
  const float* musb = mu  + (size_t)sb * KK * DD;
  const float* tasb = tau + (size_t)sb * KK * DD;

  // ---- Stage W1 / mu / tau / W2 into LDS ----
  for (int i = tid; i < 96 * HH; i += TPB) {
    const int r = i >> 7;            // /128
    const int c = i & (HH - 1);
    ldsW1[r * W1_STRIDE + c] = W1[i];
  }
  for (int i = tid; i < KK * DD; i += TPB) {     // 512 elems each
    ldsMT[i] = musb[i];
    ldsMT[KK * DD + i] = tasb[i];
  }
  if (tid < HH) ldsW2[tid] = W2[tid];
  __syncthreads();

  // ---- hk[k][h] = b1[h] + sum_d mu*Wm + tau*Wt   (h == tid; TPB == HH) ----
  {
    const float bb = b1[tid];
    #pragma unroll 4
    for (int k = 0; k < KK; ++k) {
      float acc = bb;
      #pragma unroll
      for (int d = 0; d < DD; ++d) {
        acc = __builtin_fmaf(ldsMT[k * DD + d],           ldsW1[(DD + d) * W1_STRIDE + tid], acc);
        acc = __builtin_fmaf(ldsMT[KK * DD + k * DD + d], ldsW1[(2 * DD + d) * W1_STRIDE + tid], acc);
      }
      ldsHK[k * HH + tid] = acc;
    }
  }
  // Wm/Wt are dead from here; HX overlays them after this barrier.
  __syncthreads();

  // ---- hx tile via f32 WMMA: (128 x 32) @ (32 x 128) ----
  // 4 waves x 2 strips of 16 rows; per strip, 8 column tiles of 16, each
  // accumulated over 8 chained v_wmma_f32_16x16x4_f32 (K=4, D=32).
  const int lane  = tid & 31;
  const int lhalf = (lane >> 4) & 1;   // 0: lanes 0-15, 1: lanes 16-31
  const int lrow  = lane & 15;
  const int wave  = tid >> 5;

  for (int ss = 0; ss < 2; ++ss) {
    const int strip = wave * 2 + ss;                 // 0..7
    const int m0 = strip * 16;
    const int grow = rowBase + m0 + lrow;            // row within this (s,b)
    const float* xrow = xsb + (size_t)grow * DD;

    // A fragment (16x4 f32 per step): VGPR0 = K0 / K2, VGPR1 = K1 / K3
    v2f a[8];
    #pragma unroll
    for (int t = 0; t < 8; ++t) {
      const int d = 4 * t + 2 * lhalf;
      a[t] = *reinterpret_cast<const v2f*>(xrow + d);   // global b64 load
    }

    for (int ht = 0; ht < 8; ++ht) {
      const int h0 = ht * 16;
      v8f acc = {};
      #pragma unroll
      for (int t = 0; t < 8; ++t) {
        const int d = 4 * t + 2 * lhalf;
        v2f b;
        b.x = ldsW1[d * W1_STRIDE + h0 + lrow];         // B: row K=d,   col N=lrow
        b.y = ldsW1[(d + 1) * W1_STRIDE + h0 + lrow];   //    row K=d+1
        acc = __builtin_amdgcn_wmma_f32_16x16x4_f32(
            false, a[t], false, b, (short)0, acc, false, false);
      }
      // D fragment: VGPR i -> M = i (lanes 0-15) / i+8 (lanes 16-31), N = lrow
      #pragma unroll
      for (int i = 0; i < 8; ++i) {
        const int m = m0 + i + 8 * lhalf;
        ldsHX[m * HX_STRIDE + h0 + lrow] = acc[i];
      }
    }
  }
  __syncthreads();

  // ---- gamma[n][k] = sum_h tanh(hx + hk) * W2[h]; softmax over k ----
  // One thread per row; softmax axis K is thread-local. b2 cancels in softmax.
  // h processed in pairs: all hot-loop LDS traffic is b64 (hx reads are
  // bank-conflict-free: lane stride 130 covers all 64 banks exactly once).
  float g[KK];
  #pragma unroll
  for (int k = 0; k < KK; ++k) g[k] = 0.0f;

  const float* hxRow = ldsHX + tid * HX_STRIDE;
  #pragma unroll 2
  for (int h = 0; h < HH; h += 2) {
    const v2f xv = *reinterpret_cast<const v2f*>(hxRow + h);
    const v2f wv = *reinterpret_cast<const v2f*>(ldsW2 + h);    // broadcast
    #pragma unroll
    for (int k = 0; k < KK; ++k) {
      const v2f hkv = *reinterpret_cast<const v2f*>(ldsHK + k * HH + h);  // broadcast
      g[k] = __builtin_fmaf(fast_tanh(xv.x + hkv.x), wv.x, g[k]);
      g[k] = __builtin_fmaf(fast_tanh(xv.y + hkv.y), wv.y, g[k]);
    }
  }

  // softmax over K = 16 (in registers)
  float mx = g[0];
  #pragma unroll
  for (int k = 1; k < KK; ++k) mx = fmaxf(mx, g[k]);
  float sum = 0.0f;
  #pragma unroll
  for (int k = 0; k < KK; ++k) {
    g[k] = __expf(g[k] - mx);
    sum += g[k];
  }
  const float inv = 1.0f / sum;
  #pragma unroll
  for (int k = 0; k < KK; ++k) g[k] *= inv;

  float* orow = out + ((size_t)sb * NN + rowBase + tid) * KK;   // 64B aligned
  v4f* ov = reinterpret_cast<v4f*>(orow);
  #pragma unroll
  for (int q = 0; q < 4; ++q) {
    v4f o = {g[4 * q + 0], g[4 * q + 1], g[4 * q + 2], g[4 * q + 3]};
    ov[q] = o;                                         // global b128 stores
  }
}

extern "C" void kernel_launch(void* const* d_in, const int* in_sizes, int n_in,
                              void* d_out, int out_size, void* d_ws, size_t ws_size,
                              hipStream_t stream) {
  (void)in_sizes; (void)n_in; (void)out_size; (void)d_ws; (void)ws_size;
  const float* x   = (const float*)d_in[0];
  const float* mu  = (const float*)d_in[1];
  const float* tau = (const float*)d_in[2];
  const float* W1  = (const float*)d_in[3];
  const float* b1  = (const float*)d_in[4];
  const float* W2  = (const float*)d_in[5];
  // d_in[6] (b2) intentionally unused: softmax is shift-invariant.
  float* out = (float*)d_out;

  const size_t ldsBytes = (size_t)LDS_FLOATS * sizeof(float);  // ~94 KB -> 3 WGs/WGP
  apg_fused_kernel<<<dim3(NUM_WG), dim3(TPB), ldsBytes, stream>>>(
      x, mu, tau, W1, b1, W2, out);
}